// GS_90005334655936
// MI455X (gfx1250) — compile-verified
//
#include <hip/hip_runtime.h>

typedef float v2f __attribute__((ext_vector_type(2)));
typedef float v8f __attribute__((ext_vector_type(8)));

#define LOG2E  1.44269504088896340736f
#define SMALLC 1e-8f

#if __has_builtin(__builtin_amdgcn_exp2f)
#define DEV_EXP2(x) __builtin_amdgcn_exp2f(x)
#else
#define DEV_EXP2(x) exp2f(x)
#endif

#if __has_builtin(__builtin_amdgcn_logf)
#define DEV_LOG2(x) __builtin_amdgcn_logf(x)
#else
#define DEV_LOG2(x) log2f(x)
#endif

// ---------------------------------------------------------------------------
// Kernel A: per-gaussian coefficient precompute.
// Q[k][8] = coefficients of expo*log2(e) as polynomial in [x^2, xy, y^2, x, y, 1, 0, 0]
// with log2(alpha_k) folded into the constant term so the weight is a single exp2.
// colQ[k][4] = clipped RGB (w unused).
// ---------------------------------------------------------------------------
__global__ void gs_precompute(const float* __restrict__ mu,
                              const float* __restrict__ alpha,
                              const float* __restrict__ color,
                              const float* __restrict__ scales,
                              const float* __restrict__ thetas,
                              float* __restrict__ Q,
                              float* __restrict__ colQ,
                              int K) {
  int k = blockIdx.x * blockDim.x + threadIdx.x;
  if (k >= K) return;

  float a  = fminf(fmaxf(alpha[k], 0.0f), 1.0f);
  float cr = fminf(fmaxf(color[3 * k + 0], 0.0f), 255.0f);
  float cg = fminf(fmaxf(color[3 * k + 1], 0.0f), 255.0f);
  float cb = fminf(fmaxf(color[3 * k + 2], 0.0f), 255.0f);
  float sx = fmaxf(scales[2 * k + 0], 0.1f);
  float sy = fmaxf(scales[2 * k + 1], 0.1f);
  float th = thetas[k];
  float c = cosf(th), s = sinf(th);
  float i0 = 1.0f / (sx * sx);
  float i1 = 1.0f / (sy * sy);

  // sigma_inv = R diag(i0,i1) R^T = [[A,B],[B,C]]
  float A = c * c * i0 + s * s * i1;
  float B = c * s * (i0 - i1);
  float C = s * s * i0 + c * c * i1;

  float mx = mu[2 * k + 0], my = mu[2 * k + 1];

  // expo = -0.5*(A dx^2 + 2B dx dy + C dy^2); expand around (mx,my), scale by log2(e)
  float q0 = -0.5f * A * LOG2E;
  float q1 = -B * LOG2E;
  float q2 = -0.5f * C * LOG2E;
  float q3 = (A * mx + B * my) * LOG2E;
  float q4 = (B * mx + C * my) * LOG2E;
  float q5 = -0.5f * (A * mx * mx + 2.0f * B * mx * my + C * my * my) * LOG2E
             + DEV_LOG2(a);  // fold alpha: a*exp(e) = exp2(e*log2e + log2 a); a=0 -> -inf -> 0

  float* q = Q + k * 8;
  q[0] = q0; q[1] = q1; q[2] = q2; q[3] = q3;
  q[4] = q4; q[5] = q5; q[6] = 0.0f; q[7] = 0.0f;

  float* cq = colQ + k * 4;
  cq[0] = cr; cq[1] = cg; cq[2] = cb; cq[3] = 0.0f;
}

// ---------------------------------------------------------------------------
// Kernel B: fused WMMA exponent GEMM + sequential alpha blend.
// One wave = 32 pixels (two 16-pixel WMMA N-tiles). For each 16-gaussian tile:
//   acc[16g x 16p] = Q_tile[16x8] * P[8x16] via 2x V_WMMA_F32_16X16X4_F32,
//   regather halves with shfl_xor(16) so lane L owns pixel waveBase+L,
//   then 16 faithful blend steps (eps divide + clamps) per lane.
// ---------------------------------------------------------------------------
__global__ void __launch_bounds__(256) gs_render(const float2* __restrict__ pos,
                                                 const float* __restrict__ Q,
                                                 const float* __restrict__ colQ,
                                                 float4* __restrict__ out,
                                                 int N, int K) {
  const unsigned tid      = threadIdx.x;
  const unsigned lane     = tid & 31u;
  const unsigned half     = (lane >> 4) & 1u;      // which 8-row half of the WMMA tile
  const unsigned c16      = lane & 15u;            // column within 16-wide tile
  const unsigned waveBase = blockIdx.x * blockDim.x + (tid & ~31u);
  const unsigned pix      = waveBase + lane;

  // Positions for the two 16-pixel tiles this wave owns (clamped for tail safety;
  // all lanes stay active so EXEC is all-1s at every WMMA).
  unsigned i0 = waveBase + c16;       if (i0 >= (unsigned)N) i0 = N - 1;
  unsigned i1 = waveBase + 16 + c16;  if (i1 >= (unsigned)N) i1 = N - 1;
  float2 p0 = pos[i0];
  float2 p1 = pos[i1];

  // B operand layout assumption (mirror of documented f32 A 16x4 layout):
  // VGPR v of K-chunk c holds feature 4c + v + 2*half for pixel column (lane&15).
  // features: [x^2, xy, y^2, x, y, 1, 0, 0]
  v2f b0c0, b0c1, b1c0, b1c1;
  b0c0[0] = half ? (p0.y * p0.y) : (p0.x * p0.x);
  b0c0[1] = half ? p0.x          : (p0.x * p0.y);
  b0c1[0] = half ? 0.0f          : p0.y;
  b0c1[1] = half ? 0.0f          : 1.0f;
  b1c0[0] = half ? (p1.y * p1.y) : (p1.x * p1.x);
  b1c0[1] = half ? p1.x          : (p1.x * p1.y);
  b1c1[0] = half ? 0.0f          : p1.y;
  b1c1[1] = half ? 0.0f          : 1.0f;

  float pa = 0.0f, pcx = 0.0f, pcy = 0.0f, pcz = 0.0f;

  const int ntiles = K >> 4;
  for (int t = 0; t < ntiles; ++t) {
    // A operand: lane (row m = lane&15) loads Q[g][feat], feat = 4c + v + 2*half.
    const unsigned g = (unsigned)t * 16u + c16;
    const float* qrow = Q + g * 8 + 2 * half;
    float2 qa0 = *(const float2*)(qrow);       // feats {0,1}+2h  (chunk 0)
    float2 qa1 = *(const float2*)(qrow + 4);   // feats {4,5}+2h  (chunk 1)
    v2f a0; a0[0] = qa0.x; a0[1] = qa0.y;
    v2f a1; a1[0] = qa1.x; a1[1] = qa1.y;

    v8f acc0 = {};
    acc0 = __builtin_amdgcn_wmma_f32_16x16x4_f32(false, a0, false, b0c0,
                                                 (short)0, acc0, false, false);
    acc0 = __builtin_amdgcn_wmma_f32_16x16x4_f32(false, a1, false, b0c1,
                                                 (short)0, acc0, false, false);
    v8f acc1 = {};
    acc1 = __builtin_amdgcn_wmma_f32_16x16x4_f32(false, a0, false, b1c0,
                                                 (short)0, acc1, false, false);
    acc1 = __builtin_amdgcn_wmma_f32_16x16x4_f32(false, a1, false, b1c1,
                                                 (short)0, acc1, false, false);

    // Regather: C-tile VGPR i holds row i (lanes 0-15) / row i+8 (lanes 16-31).
    // After the swap, lane L holds all 16 weights of pixel waveBase+L.
    float w[16];
#pragma unroll
    for (int i = 0; i < 8; ++i) {
      float v0 = acc0[i], v1 = acc1[i];
      float s0 = __shfl_xor(v0, 16, 32);
      float s1 = __shfl_xor(v1, 16, 32);
      float elo = half ? s1 : v0;   // gaussian row i     for my pixel
      float ehi = half ? v1 : s0;   // gaussian row i + 8 for my pixel
      w[i]     = DEV_EXP2(elo);     // = alpha_k * exp(expo)  (log2 alpha folded in)
      w[i + 8] = DEV_EXP2(ehi);
    }

    // Faithful sequential blend over the 16 gaussians of this tile.
#pragma unroll
    for (int m = 0; m < 16; ++m) {
      int k = t * 16 + m;                    // uniform across wave -> scalar loads
      float cr = colQ[4 * k + 0];
      float cg = colQ[4 * k + 1];
      float cb = colQ[4 * k + 2];
      float w1 = w[m] * (1.0f - pa);
      float na = pa + w1;
      float r  = __builtin_amdgcn_rcpf(na + SMALLC);
      pcx = fminf(fmaxf((pcx * pa + cr * w1) * r, 0.0f), 255.0f);
      pcy = fminf(fmaxf((pcy * pa + cg * w1) * r, 0.0f), 255.0f);
      pcz = fminf(fmaxf((pcz * pa + cb * w1) * r, 0.0f), 255.0f);
      pa  = fminf(na, 1.0f);                 // na >= 0 always
    }
  }

  if (pix < (unsigned)N) {
    out[pix] = make_float4(pcx, pcy, pcz, pa * 255.0f);
  }
}

// ---------------------------------------------------------------------------
extern "C" void kernel_launch(void* const* d_in, const int* in_sizes, int n_in,
                              void* d_out, int out_size, void* d_ws, size_t ws_size,
                              hipStream_t stream) {
  const float*  pos    = (const float*)d_in[0];   // [N,2]
  const float*  mu     = (const float*)d_in[1];   // [K,2]
  const float*  alpha  = (const float*)d_in[2];   // [K]
  const float*  color  = (const float*)d_in[3];   // [K,3]
  const float*  scales = (const float*)d_in[4];   // [K,2]
  const float*  thetas = (const float*)d_in[5];   // [K]

  const int N = in_sizes[0] / 2;
  const int K = in_sizes[2];

  float* Q    = (float*)d_ws;               // K*8 floats
  float* colQ = Q + (size_t)K * 8;          // K*4 floats

  gs_precompute<<<(K + 127) / 128, 128, 0, stream>>>(mu, alpha, color, scales,
                                                     thetas, Q, colQ, K);

  const int threads = 256;
  const int blocks  = (N + threads - 1) / threads;
  gs_render<<<blocks, threads, 0, stream>>>((const float2*)pos, Q, colQ,
                                            (float4*)d_out, N, K);
}